// QKVParallelLinearWithToppings_30262339567848
// MI455X (gfx1250) — compile-verified
//
#include <hip/hip_runtime.h>

typedef __attribute__((ext_vector_type(2))) float v2f;
typedef __attribute__((ext_vector_type(8))) float v8f;

#define S_TOK   4096
#define H_DIM   4096
#define R_RANK  16
#define DOUT    12288
#define DQ_DIM  4096

#define BM 128          // block rows (tokens)
#define BN 128          // block cols (qkv out)
#define BK 32           // K-tile
#define BKP 34          // padded K-major LDS stride (bank-conflict-free frag reads)

// ---------------------------------------------------------------------------
// Kernel 1: xa[t, 0:48] = x[t, :] @ A_qkv[widx[t]]   (rank-16 chunks q|k|v)
// One block per token; x row staged in LDS; 48 lanes reduce one column each.
// All adapters (6 MB) are L2-resident, so the strided A reads are cheap.
// ---------------------------------------------------------------------------
__global__ __launch_bounds__(64) void lora_xa_kernel(
    const float* __restrict__ x, const int* __restrict__ widx,
    const float* __restrict__ A, float* __restrict__ xa)
{
  __shared__ float xs[H_DIM];
  const int t = blockIdx.x;
  for (int h = threadIdx.x; h < H_DIM; h += 64)
    xs[h] = x[(size_t)t * H_DIM + h];
  __syncthreads();

  const int j = threadIdx.x;
  if (j < 3 * R_RANK) {
    const int a = widx[t];
    const float* Ap = A + (size_t)a * H_DIM * (3 * R_RANK) + j;
    float acc = 0.f;
#pragma unroll 8
    for (int h = 0; h < H_DIM; ++h)
      acc += xs[h] * Ap[(size_t)h * (3 * R_RANK)];
    xa[(size_t)t * (3 * R_RANK) + j] = acc;
  }
}

// ---------------------------------------------------------------------------
// Kernel 2: out = x @ W_base + per-token LoRA delta (rank-16 epilogue)
// 8 wave32s / block; macro-tile 128x128; per-wave 32x64 = 2x4 WMMA f32 tiles.
// Both A and B tiles are stored K-major in LDS so every WMMA fragment is one
// contiguous 8-byte LDS read into an adjacent (even) register pair — no
// register-assembly movs. Double-buffered LDS: next K-tile global loads issue
// before the WMMA block; one barrier per K-tile.
// ---------------------------------------------------------------------------
__global__ __launch_bounds__(256) void qkv_lora_gemm_kernel(
    const float* __restrict__ x, const int* __restrict__ widx,
    const float* __restrict__ W, const float* __restrict__ Bq,
    const float* __restrict__ Bkv, const float* __restrict__ xa,
    float* __restrict__ out)
{
  __shared__ float sA[2][BM * BKP];   // 2 x (128 rows x 32 K, stride 34) = 34.8 KB
  __shared__ float sB[2][BN * BKP];   // 2 x (128 cols x 32 K, stride 34) = 34.8 KB

  const int tid  = threadIdx.x;
  const int lane = tid & 31;
  const int wid  = tid >> 5;
  const int wm   = (wid >> 1) * 32;   // wave row origin: 0,32,64,96
  const int wn   = (wid & 1) * 64;    // wave col origin: 0,64
  const int bm0  = blockIdx.y * BM;
  const int bn0  = blockIdx.x * BN;

  const int half = lane >> 4;         // K-half select per ISA A/B layouts
  const int lm   = lane & 15;

  // Staging-load index precompute (each thread moves 4+4 float4 per K-tile)
  int arow[4], akq[4], brow[4], bjq[4];
#pragma unroll
  for (int q = 0; q < 4; ++q) {
    const int ia = q * 256 + tid;     // A: 1024 float4 = 128 rows x 8
    arow[q] = ia >> 3;  akq[q] = (ia & 7) << 2;
    const int ib = q * 256 + tid;     // B: 1024 float4 = 32 K-rows x 32
    brow[q] = ib >> 5;  bjq[q] = (ib & 31) << 2;
  }

  v8f c[2][4] = {};
  float4 ra[4], rb[4];

  // Prologue: fetch K-tile 0 and stage into buffer 0
#pragma unroll
  for (int q = 0; q < 4; ++q) {
    ra[q] = *(const float4*)(x + (size_t)(bm0 + arow[q]) * H_DIM + akq[q]);
    rb[q] = *(const float4*)(W + (size_t)brow[q] * DOUT + bn0 + bjq[q]);
  }
#pragma unroll
  for (int q = 0; q < 4; ++q) {
    float* da = &sA[0][arow[q] * BKP + akq[q]];
    da[0] = ra[q].x; da[1] = ra[q].y; da[2] = ra[q].z; da[3] = ra[q].w;
    float* db = &sB[0][bjq[q] * BKP + brow[q]];   // transpose: [col][k]
    db[0 * BKP] = rb[q].x; db[1 * BKP] = rb[q].y;
    db[2 * BKP] = rb[q].z; db[3 * BKP] = rb[q].w;
  }

  int p = 0;
  for (int k0 = 0; k0 < H_DIM; k0 += BK, p ^= 1) {
    const bool more = (k0 + BK) < H_DIM;
    __syncthreads();   // stage-p stores (prev iter / prologue) now visible

    if (more) {        // issue next tile's global loads before compute
      const int kn = k0 + BK;
#pragma unroll
      for (int q = 0; q < 4; ++q) {
        ra[q] = *(const float4*)(x + (size_t)(bm0 + arow[q]) * H_DIM + kn + akq[q]);
        rb[q] = *(const float4*)(W + (size_t)(kn + brow[q]) * DOUT + bn0 + bjq[q]);
      }
      if (k0 + 2 * BK < H_DIM)  // L2 prefetch one more tile ahead
        __builtin_prefetch(W + (size_t)(k0 + 2 * BK) * DOUT + bn0 + ((tid & 31) << 2), 0, 1);
    }

    // Compute on buffer p: 8 K-steps x (2x4) WMMA tiles
#pragma unroll
    for (int kk = 0; kk < BK; kk += 4) {
      const int kb = kk + 2 * half;
      v2f a0 = *(const v2f*)(&sA[p][(wm + lm)      * BKP + kb]);
      v2f a1 = *(const v2f*)(&sA[p][(wm + 16 + lm) * BKP + kb]);
      v2f b[4];
#pragma unroll
      for (int j = 0; j < 4; ++j)
        b[j] = *(const v2f*)(&sB[p][(wn + j * 16 + lm) * BKP + kb]);
#pragma unroll
      for (int j = 0; j < 4; ++j) {
        c[0][j] = __builtin_amdgcn_wmma_f32_16x16x4_f32(false, a0, false, b[j], (short)0, c[0][j], false, false);
        c[1][j] = __builtin_amdgcn_wmma_f32_16x16x4_f32(false, a1, false, b[j], (short)0, c[1][j], false, false);
      }
    }

    if (more) {        // stage next tile into the other buffer (transposed B)
#pragma unroll
      for (int q = 0; q < 4; ++q) {
        float* da = &sA[p ^ 1][arow[q] * BKP + akq[q]];
        da[0] = ra[q].x; da[1] = ra[q].y; da[2] = ra[q].z; da[3] = ra[q].w;
        float* db = &sB[p ^ 1][bjq[q] * BKP + brow[q]];
        db[0 * BKP] = rb[q].x; db[1 * BKP] = rb[q].y;
        db[2 * BKP] = rb[q].z; db[3 * BKP] = rb[q].w;
      }
    }
  }

  // Epilogue: out = base + (xa[t, chunk] @ B_sel[idx[t]]), rank 16.
  // C layout: VGPR v -> row (v + 8*half), lane%16 -> col. Every 16-col tile
  // lies entirely in one of the q|k|v segments (all boundaries % 16 == 0);
  // k and v both map to B_kv column (gcol - 4096).
#pragma unroll
  for (int j = 0; j < 4; ++j) {
    const int gcol = bn0 + wn + j * 16 + lm;
    const float* Bbase;
    int bcol, stride, cb;
    if (gcol < DQ_DIM)      { Bbase = Bq;  bcol = gcol;          stride = DQ_DIM;     cb = 0; }
    else                    { Bbase = Bkv; bcol = gcol - DQ_DIM; stride = 2 * DQ_DIM;
                              cb = (gcol < 2 * DQ_DIM) ? 1 : 2; }
#pragma unroll
    for (int i = 0; i < 2; ++i) {
#pragma unroll
      for (int v = 0; v < 8; ++v) {
        const int grow = bm0 + wm + i * 16 + v + 8 * half;
        const int a    = widx[grow];
        const float* xr = xa + (size_t)grow * (3 * R_RANK) + cb * R_RANK;
        const float* br = Bbase + (size_t)a * R_RANK * stride + bcol;
        float d = 0.f;
#pragma unroll
        for (int r = 0; r < R_RANK; ++r)
          d += xr[r] * br[(size_t)r * stride];
        out[(size_t)grow * DOUT + gcol] = c[i][j][v] + d;
      }
    }
  }
}

// ---------------------------------------------------------------------------
extern "C" void kernel_launch(void* const* d_in, const int* in_sizes, int n_in,
                              void* d_out, int out_size, void* d_ws, size_t ws_size,
                              hipStream_t stream) {
  const float* x    = (const float*)d_in[0];
  const int*   widx = (const int*)  d_in[1];
  const float* W    = (const float*)d_in[2];
  const float* A    = (const float*)d_in[3];
  const float* Bq   = (const float*)d_in[4];
  const float* Bkv  = (const float*)d_in[5];
  float* out = (float*)d_out;
  float* xa  = (float*)d_ws;   // S * 48 floats = 768 KB scratch

  lora_xa_kernel<<<S_TOK, 64, 0, stream>>>(x, widx, A, xa);

  dim3 grid(DOUT / BN, S_TOK / BM);   // 96 x 32 blocks
  qkv_lora_gemm_kernel<<<grid, 256, 0, stream>>>(x, widx, W, Bq, Bkv, xa, out);
}